// SpeechEmbedder_35416300322958
// MI455X (gfx1250) — compile-verified
//
#include <hip/hip_runtime.h>
#include <hip/hip_bf16.h>

// Problem dims (fixed by the reference)
#define BB   16
#define TT   1024
#define CC   1024
#define TP2  (TT + 2)      // 1026
#define LNEPS 1e-5f

// Native clang vector types (usable with builtins, unlike HIP_vector_type)
typedef __attribute__((ext_vector_type(4))) int   v4i;
typedef __attribute__((ext_vector_type(4))) float v4f;
typedef __attribute__((address_space(1))) v4i* gas_v4i_ptr;  // global
typedef __attribute__((address_space(3))) v4i* las_v4i_ptr;  // LDS

// ---------------------------------------------------------------------------
// CDNA5 async global->LDS copy of 16 bytes per lane (ASYNCcnt-tracked).
// ---------------------------------------------------------------------------
__device__ __forceinline__ void async_copy16(const float* gsrc, float* ldst) {
#if __has_builtin(__builtin_amdgcn_global_load_async_to_lds_b128)
  __builtin_amdgcn_global_load_async_to_lds_b128(
      (gas_v4i_ptr)(void*)(const_cast<float*>(gsrc)),
      (las_v4i_ptr)(void*)ldst,
      0, 0);
#else
  unsigned loff = (unsigned)(unsigned long long)ldst;
  unsigned long long ga = (unsigned long long)gsrc;
  asm volatile("global_load_async_to_lds_b128 %0, %1, off"
               :: "v"(loff), "v"(ga) : "memory");
#endif
}

__device__ __forceinline__ void wait_async0() {
#if __has_builtin(__builtin_amdgcn_s_wait_asynccnt)
  __builtin_amdgcn_s_wait_asynccnt(0);
#else
  asm volatile("s_wait_asynccnt 0" ::: "memory");
#endif
}

// ---------------------------------------------------------------------------
// wave32 + 8-wave block reduction -> mean and rsqrt(var+eps) over CC elems.
// sred must hold >= 18 floats of LDS.
// ---------------------------------------------------------------------------
__device__ __forceinline__ void block_stats(float s, float ss, float* sred,
                                            float& mu, float& rstd) {
  #pragma unroll
  for (int o = 16; o > 0; o >>= 1) {
    s  += __shfl_xor(s,  o, 32);
    ss += __shfl_xor(ss, o, 32);
  }
  const int wave = threadIdx.x >> 5;   // 8 waves of 32 (wave32)
  const int lane = threadIdx.x & 31;
  if (lane == 0) { sred[wave] = s; sred[8 + wave] = ss; }
  __syncthreads();
  if (threadIdx.x == 0) {
    float S = 0.f, SS = 0.f;
    #pragma unroll
    for (int i = 0; i < 8; ++i) { S += sred[i]; SS += sred[8 + i]; }
    const float m = S * (1.0f / CC);
    const float v = SS * (1.0f / CC) - m * m;
    sred[16] = m;
    sred[17] = rsqrtf(v + LNEPS);
  }
  __syncthreads();
  mu   = sred[16];
  rstd = sred[17];
}

// ---------------------------------------------------------------------------
// Main fused kernel: one workgroup (256 thr / 8 waves) per output row.
// Async-stage the value row and the positional row into LDS, then LN.
// Also emits the pad-mask element for this row (and l2[b] from the t==0 row).
// ---------------------------------------------------------------------------
__global__ __launch_bounds__(256) void embed_main_kernel(
    const float* __restrict__ x,       const int* __restrict__ lengths,
    const float* __restrict__ pos_tab, const float* __restrict__ scale,
    const float* __restrict__ gamma,   const float* __restrict__ beta,
    const float* __restrict__ bos_ln,  const float* __restrict__ eos_ln,
    float* __restrict__ out_x,         float* __restrict__ out_pad,
    float* __restrict__ out_l2) {
  __shared__ float s_pe[CC];
  __shared__ float s_v[CC];
  __shared__ float sred[18];

  const int row = blockIdx.x;          // 0 .. B*(T+2)-1
  const int b   = row / TP2;
  const int t   = row - b * TP2;
  const int len = lengths[b];          // uniform -> scalar load
  const int l1  = len + 1;
  const int l2  = len + 2;

  const int e = threadIdx.x * 4;       // 4 floats (16B) per lane

  // fairseq make_positions collapses to: pos = t+2 if nonpad else PAD_IDX(=1)
  const int pos = (t < l2) ? (t + 2) : 1;
  async_copy16(pos_tab + (size_t)pos * CC + e, &s_pe[e]);

  // Source-vector select (block-uniform branch -> full EXEC inside)
  bool vzero = false;
  if (t == 0) {
    async_copy16(bos_ln + e, &s_v[e]);
  } else if (t == l1) {
    async_copy16(eos_ln + e, &s_v[e]);
  } else if (t <= len) {
    async_copy16(x + ((size_t)b * TT + (t - 1)) * CC + e, &s_v[e]);
  } else {
    vzero = true;                      // masked frames / zero slot
  }

  // Side outputs while the DMA is in flight
  if (threadIdx.x == 0) {
    out_pad[row] = (t >= l2) ? 1.0f : 0.0f;
    if (t == 0) out_l2[b] = (float)l2;
  }
  const float sc = scale[0];

  wait_async0();                       // drain this wave's ASYNCcnt
  __syncthreads();                     // publish LDS to all 8 waves

  const v4f pe4 = *(const v4f*)&s_pe[e];
  v4f v4 = (v4f)(0.f);
  if (!vzero) v4 = *(const v4f*)&s_v[e];

  v4f tv;
  tv.x = fmaf(v4.x, sc, pe4.x);
  tv.y = fmaf(v4.y, sc, pe4.y);
  tv.z = fmaf(v4.z, sc, pe4.z);
  tv.w = fmaf(v4.w, sc, pe4.w);

  float s  = tv.x + tv.y + tv.z + tv.w;
  float ss = fmaf(tv.x, tv.x, fmaf(tv.y, tv.y, fmaf(tv.z, tv.z, tv.w * tv.w)));
  float mu, rstd;
  block_stats(s, ss, sred, mu, rstd);

  const v4f g4 = *(const v4f*)(gamma + e);
  const v4f b4 = *(const v4f*)(beta + e);
  v4f o;
  o.x = fmaf((tv.x - mu) * rstd, g4.x, b4.x);
  o.y = fmaf((tv.y - mu) * rstd, g4.y, b4.y);
  o.z = fmaf((tv.z - mu) * rstd, g4.z, b4.z);
  o.w = fmaf((tv.w - mu) * rstd, g4.w, b4.w);
  // Write-once streaming output: non-temporal store (th:TH_STORE_NT),
  // keeps pos_table + gamma/beta resident in L2.
  __builtin_nontemporal_store(o, (v4f*)(out_x + (size_t)row * CC + e));
}

// ---------------------------------------------------------------------------
// LayerNorm the BOS / EOS embeddings into workspace (2 blocks).
// ---------------------------------------------------------------------------
__global__ __launch_bounds__(256) void ln_special_kernel(
    const float* __restrict__ bos, const float* __restrict__ eos,
    const float* __restrict__ g,   const float* __restrict__ bt,
    float* __restrict__ ws) {
  __shared__ float sred[18];
  const float* src = (blockIdx.x == 0) ? bos : eos;
  float* dst = ws + (size_t)blockIdx.x * CC;
  const int e = threadIdx.x * 4;

  v4f v = *(const v4f*)(src + e);
  float s  = v.x + v.y + v.z + v.w;
  float ss = v.x*v.x + v.y*v.y + v.z*v.z + v.w*v.w;
  float mu, rstd;
  block_stats(s, ss, sred, mu, rstd);

  v4f g4 = *(const v4f*)(g + e);
  v4f b4 = *(const v4f*)(bt + e);
  v4f o;
  o.x = fmaf((v.x - mu) * rstd, g4.x, b4.x);
  o.y = fmaf((v.y - mu) * rstd, g4.y, b4.y);
  o.z = fmaf((v.z - mu) * rstd, g4.z, b4.z);
  o.w = fmaf((v.w - mu) * rstd, g4.w, b4.w);
  *(v4f*)(dst + e) = o;
}

// ---------------------------------------------------------------------------
extern "C" void kernel_launch(void* const* d_in, const int* in_sizes, int n_in,
                              void* d_out, int out_size, void* d_ws, size_t ws_size,
                              hipStream_t stream) {
  (void)in_sizes; (void)n_in; (void)out_size; (void)ws_size;

  const float* x        = (const float*)d_in[0];
  const int*   lengths  = (const int*)  d_in[1];
  const float* bos      = (const float*)d_in[2];
  const float* eos      = (const float*)d_in[3];
  const float* ln_s_g   = (const float*)d_in[4];
  const float* ln_s_b   = (const float*)d_in[5];
  const float* pos_tab  = (const float*)d_in[6];
  const float* scale    = (const float*)d_in[7];
  const float* ln_e_g   = (const float*)d_in[8];
  const float* ln_e_b   = (const float*)d_in[9];

  float* out     = (float*)d_out;
  float* out_x   = out;                                  // (B, T+2, C) f32
  float* out_pad = out + (size_t)BB * TP2 * CC;          // (B, T+2) mask
  float* out_l2  = out_pad + (size_t)BB * TP2;           // (B,) lengths+2

  float* ws     = (float*)d_ws;
  float* bos_ln = ws;          // 1024 floats
  float* eos_ln = ws + CC;     // 1024 floats

  ln_special_kernel<<<2, 256, 0, stream>>>(bos, eos, ln_s_g, ln_s_b, ws);
  embed_main_kernel<<<BB * TP2, 256, 0, stream>>>(
      x, lengths, pos_tab, scale, ln_e_g, ln_e_b, bos_ln, eos_ln,
      out_x, out_pad, out_l2);
}